// DGL_GAT_1752346657313
// MI455X (gfx1250) — compile-verified
//
#include <hip/hip_runtime.h>
#include <cstdint>
#include <cstddef>

#define IN_F 128
#define NEG_SLOPE 0.2f
#define H0 4
#define H1 1

typedef __attribute__((ext_vector_type(2))) float v2f;
typedef __attribute__((ext_vector_type(8))) float v8f;

// ---------------------------------------------------------------------------
// Y[nrows x NCOL] = X[nrows x 128] * W[128 x NCOL], fp32 via v_wmma_f32_16x16x4_f32
// One wave per 16x16 output tile; NCOL/16 waves per block; grid.x = nrows/16.
// A-tile staged in LDS with padded stride 132 (=> 16 distinct banks for the
// 16-lane column reads). B read straight from global (tiny, L2-resident).
// ---------------------------------------------------------------------------
template <int NCOL>
__global__ __launch_bounds__(32 * (NCOL / 16)) void wmma_gemm_k128(
    const float* __restrict__ X, const float* __restrict__ W, float* __restrict__ Y) {
  constexpr int K = 128;
  __shared__ float xs[16][K + 4];  // pad: bank = (r*132 + k) % 64 unique per r
  const int row0 = blockIdx.x * 16;
  const int tid = threadIdx.x;
  const int lane = tid & 31;
  const int wave = tid >> 5;

  for (int i = tid; i < 16 * K; i += 32 * (NCOL / 16)) {
    int r = i >> 7;
    int c = i & 127;
    xs[r][c] = X[(row0 + r) * K + c];
  }
  __syncthreads();

  const int n0 = wave * 16;
  const int m = lane & 15;      // output row within tile (A row)
  const int khalf = lane >> 4;  // 0: K pair {0,1}; 1: K pair {2,3}

  v8f acc = {};
  for (int k0 = 0; k0 < K; k0 += 4) {
    const int ka = k0 + khalf * 2;
    v2f a, b;
    a.x = xs[m][ka];
    a.y = xs[m][ka + 1];
    b.x = W[(ka) *NCOL + n0 + m];
    b.y = W[(ka + 1) * NCOL + n0 + m];
    acc = __builtin_amdgcn_wmma_f32_16x16x4_f32(false, a, false, b, (short)0, acc,
                                                false, false);
  }

  // C/D layout: VGPR r -> M = r + 8*khalf, N = lane&15
  const int n = n0 + m;
#pragma unroll
  for (int r = 0; r < 8; ++r) {
    Y[(row0 + khalf * 8 + r) * NCOL + n] = acc[r];
  }
}

// ---------------------------------------------------------------------------
// el[n,h] = sum_d h[n,h,d]*al[h,d]; er likewise. D == 32 == wave32: lane = d.
// One wave per (node, head). h is [N, H*32] flat; gw = node*H + head.
// ---------------------------------------------------------------------------
template <int H>
__global__ void attn_coef32(const float* __restrict__ h, const float* __restrict__ al,
                            const float* __restrict__ ar, float* __restrict__ el,
                            float* __restrict__ er, int nh) {
  const int gw = blockIdx.x * (blockDim.x >> 5) + (threadIdx.x >> 5);
  if (gw >= nh) return;
  const int lane = threadIdx.x & 31;
  const int head = gw % H;
  const float v = h[gw * 32 + lane];
  float l = v * al[head * 32 + lane];
  float r = v * ar[head * 32 + lane];
#pragma unroll
  for (int off = 16; off > 0; off >>= 1) {
    l += __shfl_down(l, off, 32);
    r += __shfl_down(r, off, 32);
  }
  if (lane == 0) {
    el[gw] = l;
    er[gw] = r;
  }
}

// ---------------------------------------------------------------------------
// Per (edge, head): e = leaky_relu(el[src]+er[dst]); segment max into m[dst,h]
// via int/uint atomic ordering trick (m init'd to 0xFFFFFFFF).
// ---------------------------------------------------------------------------
template <int H>
__global__ void edge_e_max(const int* __restrict__ src, const int* __restrict__ dst,
                           const float* __restrict__ el, const float* __restrict__ er,
                           float* __restrict__ e, float* __restrict__ m, int ne) {
  const int i = blockIdx.x * blockDim.x + threadIdx.x;
  if (i >= ne * H) return;
  const int eid = i / H;
  const int h = i % H;
  const int d = dst[eid];
  float v = el[src[eid] * H + h] + er[d * H + h];
  v = (v > 0.0f) ? v : NEG_SLOPE * v;
  e[i] = v;
  float* slot = m + d * H + h;
  if (v >= 0.0f)
    atomicMax((int*)slot, __float_as_int(v));
  else
    atomicMin((unsigned int*)slot, __float_as_uint(v));
}

// ---------------------------------------------------------------------------
// ex = exp(e - m[dst]); in-place over e; segment sum into s[dst,h].
// ---------------------------------------------------------------------------
template <int H>
__global__ void edge_ex_sum(const int* __restrict__ dst, const float* __restrict__ m,
                            float* __restrict__ e, float* __restrict__ s, int ne) {
  const int i = blockIdx.x * blockDim.x + threadIdx.x;
  if (i >= ne * H) return;
  const int eid = i / H;
  const int h = i % H;
  const int slot = dst[eid] * H + h;
  const float ex = expf(e[i] - m[slot]);
  e[i] = ex;
  atomicAdd(s + slot, ex);
}

// ---------------------------------------------------------------------------
// out[dst, h, d] += (ex[eid,h]/s[dst,h]) * hfeat[src, h, d].  H*D threads/edge.
// ---------------------------------------------------------------------------
template <int H, int D>
__global__ void edge_aggregate(const int* __restrict__ src, const int* __restrict__ dst,
                               const float* __restrict__ ex, const float* __restrict__ s,
                               const float* __restrict__ hfeat, float* __restrict__ out,
                               int ne) {
  constexpr int HD = H * D;
  const long long t = (long long)blockIdx.x * blockDim.x + threadIdx.x;
  if (t >= (long long)ne * HD) return;
  const int eid = (int)(t / HD);
  const int c = (int)(t % HD);
  const int h = c / D;
  const int sn = src[eid];
  const int dn = dst[eid];
  const float alpha = ex[eid * H + h] / s[dn * H + h];
  atomicAdd(out + dn * HD + c, alpha * hfeat[sn * HD + c]);
}

// act = elu(agg + b), C must be power of two
template <int C>
__global__ void bias_elu(const float* __restrict__ agg, const float* __restrict__ b,
                         float* __restrict__ act, int total) {
  const int i = blockIdx.x * blockDim.x + threadIdx.x;
  if (i >= total) return;
  const float v = agg[i] + b[i & (C - 1)];
  act[i] = (v > 0.0f) ? v : (expf(v) - 1.0f);
}

__global__ void add_bias32(float* __restrict__ out, const float* __restrict__ b,
                           int total) {
  const int i = blockIdx.x * blockDim.x + threadIdx.x;
  if (i >= total) return;
  out[i] += b[i & 31];
}

extern "C" void kernel_launch(void* const* d_in, const int* in_sizes, int n_in,
                              void* d_out, int out_size, void* d_ws, size_t ws_size,
                              hipStream_t stream) {
  const float* x = (const float*)d_in[0];
  const int* src = (const int*)d_in[1];
  const int* dst = (const int*)d_in[2];
  const float* W1 = (const float*)d_in[3];
  const float* al1 = (const float*)d_in[4];
  const float* ar1 = (const float*)d_in[5];
  const float* b1 = (const float*)d_in[6];
  const float* W2 = (const float*)d_in[7];
  const float* al2 = (const float*)d_in[8];
  const float* ar2 = (const float*)d_in[9];
  const float* b2 = (const float*)d_in[10];

  const int n = in_sizes[0] / IN_F;  // 100000 (divisible by 16)
  const int e = in_sizes[1];         // 1600000

  char* ws = (char*)d_ws;
  size_t off = 0;
  auto wsalloc = [&](size_t bytes) {
    void* p = ws + off;
    off = (off + bytes + 255) & ~(size_t)255;
    return p;
  };
  float* h1 = (float*)wsalloc((size_t)n * 128 * sizeof(float));    // h1, then act
  float* agg1 = (float*)wsalloc((size_t)n * 128 * sizeof(float));  // agg1, then g2
  float* ebuf = (float*)wsalloc((size_t)e * H0 * sizeof(float));   // e / ex (both layers)
  float* el = (float*)wsalloc((size_t)n * H0 * sizeof(float));
  float* er = (float*)wsalloc((size_t)n * H0 * sizeof(float));
  float* mbuf = (float*)wsalloc((size_t)n * H0 * sizeof(float));
  float* sbuf = (float*)wsalloc((size_t)n * H0 * sizeof(float));
  float* outv = (float*)d_out;

  // ======================= Layer 1 (H=4, D=32) ==========================
  wmma_gemm_k128<128><<<n / 16, 256, 0, stream>>>(x, W1, h1);

  {
    const int nh = n * H0;
    attn_coef32<H0><<<(nh + 7) / 8, 256, 0, stream>>>(h1, al1, ar1, el, er, nh);
  }

  hipMemsetAsync(mbuf, 0xFF, (size_t)n * H0 * sizeof(float), stream);
  hipMemsetAsync(sbuf, 0, (size_t)n * H0 * sizeof(float), stream);
  hipMemsetAsync(agg1, 0, (size_t)n * 128 * sizeof(float), stream);

  edge_e_max<H0><<<(e * H0 + 255) / 256, 256, 0, stream>>>(src, dst, el, er, ebuf, mbuf, e);
  edge_ex_sum<H0><<<(e * H0 + 255) / 256, 256, 0, stream>>>(dst, mbuf, ebuf, sbuf, e);
  {
    const long long total = (long long)e * 128;
    edge_aggregate<H0, 32><<<(int)((total + 255) / 256), 256, 0, stream>>>(
        src, dst, ebuf, sbuf, h1, agg1, e);
  }

  // act = elu(agg1 + b1), overwriting h1 (dead after aggregation)
  bias_elu<128><<<(n * 128 + 255) / 256, 256, 0, stream>>>(agg1, b1, h1, n * 128);

  // ======================= Layer 2 (H=1, D=32) ==========================
  float* g2 = agg1;  // agg1 dead after bias_elu; reuse for projection output
  wmma_gemm_k128<32><<<n / 16, 64, 0, stream>>>(h1, W2, g2);

  attn_coef32<H1><<<(n + 7) / 8, 256, 0, stream>>>(g2, al2, ar2, el, er, n);

  hipMemsetAsync(mbuf, 0xFF, (size_t)n * H1 * sizeof(float), stream);
  hipMemsetAsync(sbuf, 0, (size_t)n * H1 * sizeof(float), stream);
  hipMemsetAsync(d_out, 0, (size_t)n * 32 * sizeof(float), stream);

  edge_e_max<H1><<<(e + 255) / 256, 256, 0, stream>>>(src, dst, el, er, ebuf, mbuf, e);
  edge_ex_sum<H1><<<(e + 255) / 256, 256, 0, stream>>>(dst, mbuf, ebuf, sbuf, e);
  {
    const long long total = (long long)e * 32;
    edge_aggregate<H1, 32><<<(int)((total + 255) / 256), 256, 0, stream>>>(
        src, dst, ebuf, sbuf, g2, outv, e);
  }

  // final: out = agg2 + b2   (mean over the single head == identity)
  add_bias32<<<(n * 32 + 255) / 256, 256, 0, stream>>>(outv, b2, n * 32);
}